// TDSDecoder_56049323213346
// MI455X (gfx1250) — compile-verified
//
#include <hip/hip_runtime.h>
#include <hip/hip_bf16.h>

typedef __attribute__((ext_vector_type(2))) float v2f;
typedef __attribute__((ext_vector_type(8))) float v8f;

#define HID      512
#define TENC     16384
#define VOCAB    31
#define MAXLEN   100
#define EOSIDX   29

// workspace layout (float offsets)
#define H_OFF    0
#define Y_OFF    512
#define SC_OFF   1024
#define AT_OFF   (1024 + TENC)
#define SUM_OFF  (AT_OFF + TENC)
#define MASK_OFF (SUM_OFF + 512)
#define CTRL_OFF (MASK_OFF + TENC)

// out layout: logits [100*31] | length [1] | attn [100*16384]
#define LEN_OFF_OUT  3100
#define ATTN_OFF_OUT 3101

__global__ void __launch_bounds__(512) init_kernel(const float* __restrict__ embed,
                                                   float* __restrict__ ws) {
  int i = threadIdx.x;
  ws[H_OFF + i] = 0.0f;
  ws[Y_OFF + i] = embed[EOSIDX * HID + i];
  if (i == 0) {
    int* ctrl = (int*)(ws + CTRL_OFF);
    ctrl[0] = MAXLEN;  // length
    ctrl[1] = 0;       // eos-found flag
  }
}

__global__ void __launch_bounds__(256) mask_kernel(const float* __restrict__ enc,
                                                   float* __restrict__ ws) {
  int t = blockIdx.x * 256 + threadIdx.x;
  if (t >= TENC) return;
  const float* row = enc + (size_t)t * 1024;  // keys[t][*]
  int m = 0;
  for (int j = 0; j < HID; ++j) {
    if (row[j] != 30.0f) { m = 1; break; }
  }
  ((int*)(ws + MASK_OFF))[t] = m;
}

__global__ void __launch_bounds__(512) gru_kernel(const float* __restrict__ wih,
                                                  const float* __restrict__ whh,
                                                  const float* __restrict__ bih,
                                                  const float* __restrict__ bhh,
                                                  float* __restrict__ ws) {
  __shared__ float ys[HID];
  __shared__ float hs[HID];
  int i = threadIdx.x;
  ys[i] = ws[Y_OFF + i];
  hs[i] = ws[H_OFF + i];
  __syncthreads();
  const float* wr = wih + (size_t)i * HID;
  const float* wz = wih + (size_t)(HID + i) * HID;
  const float* wn = wih + (size_t)(2 * HID + i) * HID;
  const float* ur = whh + (size_t)i * HID;
  const float* uz = whh + (size_t)(HID + i) * HID;
  const float* un = whh + (size_t)(2 * HID + i) * HID;
  float gxr = bih[i], gxz = bih[HID + i], gxn = bih[2 * HID + i];
  float ghr = bhh[i], ghz = bhh[HID + i], ghn = bhh[2 * HID + i];
  for (int j = 0; j < HID; ++j) {
    float yj = ys[j], hj = hs[j];
    gxr = fmaf(yj, wr[j], gxr);
    gxz = fmaf(yj, wz[j], gxz);
    gxn = fmaf(yj, wn[j], gxn);
    ghr = fmaf(hj, ur[j], ghr);
    ghz = fmaf(hj, uz[j], ghz);
    ghn = fmaf(hj, un[j], ghn);
  }
  float r = 1.0f / (1.0f + expf(-(gxr + ghr)));
  float z = 1.0f / (1.0f + expf(-(gxz + ghz)));
  float n = tanhf(gxn + r * ghn);
  ws[H_OFF + i] = (1.0f - z) * n + z * hs[i];
}

// scores[t0+n] = (h . keys[t0+n]) * scale via V_WMMA_F32_16X16X4_F32.
// A: row M=0 = h chunk (lane0 -> K=0,1 ; lane16 -> K=2,3), other rows 0
//    (built branchlessly: all lanes load, v_cndmask keeps only N==0 lanes).
// B: 4x16 tile of keys^T: lane L -> N=L&15, K = (L<16?0:2)+{0,1}.
__global__ void __launch_bounds__(256) scores_kernel(const float* __restrict__ enc,
                                                     float* __restrict__ ws) {
  const float* h = ws + H_OFF;
  const int* mask = (const int*)(ws + MASK_OFF);
  int t0 = blockIdx.x * 16;
  int lane = threadIdx.x & 31;
  int wave = threadIdx.x >> 5;     // 0..7; wave covers k in [wave*64, wave*64+64)
  int n = lane & 15;
  int koff = (lane >> 4) * 2;      // 0 or 2
  bool sel = (n == 0);
  const float* keyrow = enc + (size_t)(t0 + n) * 1024;  // keys[t0+n][*]

  auto step = [&](int k0, v8f c) -> v8f {
    float2 bf = *(const float2*)(keyrow + k0 + koff);
    float2 af = *(const float2*)(h + k0 + koff);      // broadcast load, all lanes
    v2f a; a[0] = sel ? af.x : 0.0f; a[1] = sel ? af.y : 0.0f;
    v2f b; b[0] = bf.x; b[1] = bf.y;
    return __builtin_amdgcn_wmma_f32_16x16x4_f32(false, a, false, b, (short)0, c,
                                                 false, false);
  };

  v8f c0 = {}, c1 = {}, c2 = {}, c3 = {};
  int kbase = wave * 64;
#pragma unroll
  for (int i = 0; i < 4; ++i) {
    int k = kbase + i * 16;        // 4 independent chains of depth 4
    c0 = step(k,      c0);
    c1 = step(k + 4,  c1);
    c2 = step(k + 8,  c2);
    c3 = step(k + 12, c3);
  }
  float partial = c0[0] + c1[0] + c2[0] + c3[0];  // row M=0: VGPR0, lanes 0..15
  __shared__ float red[8][16];
  if (lane < 16) red[wave][lane] = partial;
  __syncthreads();
  if (threadIdx.x < 16) {
    float s = 0.0f;
    for (int w = 0; w < 8; ++w) s += red[w][threadIdx.x];
    s *= 0.04419417382415922f;     // 1/sqrt(512)
    int key = t0 + threadIdx.x;
    (ws + SC_OFF)[key] = mask[key] ? s : -1.0e9f;
  }
}

__global__ void __launch_bounds__(1024) softmax_kernel(float* __restrict__ ws,
                                                       float* __restrict__ out, int t) {
  __shared__ float red[1024];
  int tid = threadIdx.x;
  const float* sc = ws + SC_OFF;
  float v[16];
  float m = -3.0e38f;
  for (int j = 0; j < 16; ++j) {
    v[j] = sc[tid + j * 1024];
    m = fmaxf(m, v[j]);
  }
  red[tid] = m; __syncthreads();
  for (int s = 512; s > 0; s >>= 1) {
    if (tid < s) red[tid] = fmaxf(red[tid], red[tid + s]);
    __syncthreads();
  }
  m = red[0]; __syncthreads();
  float sum = 0.0f;
  for (int j = 0; j < 16; ++j) { v[j] = expf(v[j] - m); sum += v[j]; }
  red[tid] = sum; __syncthreads();
  for (int s = 512; s > 0; s >>= 1) {
    if (tid < s) red[tid] += red[tid + s];
    __syncthreads();
  }
  float inv = 1.0f / red[0];
  float* attn = ws + AT_OFF;
  float* oat = out + ATTN_OFF_OUT + (size_t)t * TENC;
  for (int j = 0; j < 16; ++j) {
    float a = v[j] * inv;
    attn[tid + j * 1024] = a;
    oat[tid + j * 1024] = a;
  }
}

// summary[n0+n] = sum_k attn[k] * values[k][n0+n] via WMMA f32 16x16x4.
__global__ void __launch_bounds__(256) summary_kernel(const float* __restrict__ enc,
                                                      float* __restrict__ ws) {
  const float* attn = ws + AT_OFF;
  int n0 = blockIdx.x * 16;
  int lane = threadIdx.x & 31;
  int wave = threadIdx.x >> 5;     // wave covers k in [wave*2048, +2048)
  int n = lane & 15;
  int koff = (lane >> 4) * 2;
  bool sel = (n == 0);

  auto step = [&](int k0, v8f c) -> v8f {
    const float* p = enc + (size_t)(k0 + koff) * 1024 + 512 + n0 + n;  // values
    v2f b; b[0] = p[0]; b[1] = p[1024];
    float2 af = *(const float2*)(attn + k0 + koff);   // broadcast load
    v2f a; a[0] = sel ? af.x : 0.0f; a[1] = sel ? af.y : 0.0f;
    return __builtin_amdgcn_wmma_f32_16x16x4_f32(false, a, false, b, (short)0, c,
                                                 false, false);
  };

  v8f c0 = {}, c1 = {}, c2 = {}, c3 = {};
  int kw = wave * 2048;
  for (int s = 0; s < 512; s += 4) {
    int k = kw + s * 4;            // 4 independent chains
    c0 = step(k,      c0);
    c1 = step(k + 4,  c1);
    c2 = step(k + 8,  c2);
    c3 = step(k + 12, c3);
  }
  float partial = c0[0] + c1[0] + c2[0] + c3[0];
  __shared__ float red[8][16];
  if (lane < 16) red[wave][lane] = partial;
  __syncthreads();
  if (threadIdx.x < 16) {
    float sum = 0.0f;
    for (int w = 0; w < 8; ++w) sum += red[w][threadIdx.x];
    ws[SUM_OFF + n0 + threadIdx.x] = sum;
  }
}

__global__ void __launch_bounds__(32) logits_kernel(const float* __restrict__ wout,
                                                    const float* __restrict__ bout,
                                                    const float* __restrict__ embed,
                                                    float* __restrict__ ws,
                                                    float* __restrict__ out, int t) {
  __shared__ float lg[32];
  __shared__ int bestS;
  int lane = threadIdx.x;
  float acc = -3.0e38f;
  if (lane < VOCAB) {
    const float* w = wout + (size_t)lane * 1024;
    const float* sm = ws + SUM_OFF;
    const float* h = ws + H_OFF;
    acc = bout[lane];
    for (int j = 0; j < HID; ++j) acc = fmaf(sm[j], w[j], acc);
    for (int j = 0; j < HID; ++j) acc = fmaf(h[j], w[HID + j], acc);
    out[t * VOCAB + lane] = acc;
  }
  lg[lane] = acc;
  __syncthreads();
  if (lane == 0) {
    int best = 0; float bv = lg[0];
    for (int v = 1; v < VOCAB; ++v)
      if (lg[v] > bv) { bv = lg[v]; best = v; }   // first-index tie-break
    bestS = best;
    int* ctrl = (int*)(ws + CTRL_OFF);
    if (ctrl[1] == 0 && best == EOSIDX) { ctrl[0] = t; ctrl[1] = 1; }
  }
  __syncthreads();
  int best = bestS;
  for (int j = lane; j < HID; j += 32)
    ws[Y_OFF + j] = embed[(size_t)best * HID + j];
}

__global__ void finalize_kernel(float* __restrict__ ws, float* __restrict__ out) {
  ((int*)out)[LEN_OFF_OUT] = ((int*)(ws + CTRL_OFF))[0];  // int32 bit pattern
}

extern "C" void kernel_launch(void* const* d_in, const int* in_sizes, int n_in,
                              void* d_out, int out_size, void* d_ws, size_t ws_size,
                              hipStream_t stream) {
  const float* enc   = (const float*)d_in[0];
  // d_in[1] = encoding_lens (int64) — unused by the reference math
  const float* embed = (const float*)d_in[2];
  const float* wih   = (const float*)d_in[3];
  const float* whh   = (const float*)d_in[4];
  const float* bih   = (const float*)d_in[5];
  const float* bhh   = (const float*)d_in[6];
  const float* wout  = (const float*)d_in[7];
  const float* bout  = (const float*)d_in[8];
  float* out = (float*)d_out;
  float* ws  = (float*)d_ws;

  init_kernel<<<1, 512, 0, stream>>>(embed, ws);
  mask_kernel<<<TENC / 256, 256, 0, stream>>>(enc, ws);

  for (int t = 0; t < MAXLEN; ++t) {
    gru_kernel<<<1, 512, 0, stream>>>(wih, whh, bih, bhh, ws);
    scores_kernel<<<TENC / 16, 256, 0, stream>>>(enc, ws);
    softmax_kernel<<<1, 1024, 0, stream>>>(ws, out, t);
    summary_kernel<<<HID / 16, 256, 0, stream>>>(enc, ws);
    logits_kernel<<<1, 32, 0, stream>>>(wout, bout, embed, ws, out, t);
  }
  finalize_kernel<<<1, 1, 0, stream>>>(ws, out);
}